// MTPhysicsLoss_1073741824743
// MI455X (gfx1250) — compile-verified
//
#include <hip/hip_runtime.h>

#define NZc   256
#define NFc   16384
#define BLK   128
#define MUc   1.25663706143591729e-6f   // 4*pi*1e-7
#define TWO_PI 6.28318530717958647692f
#define RAD2DEG 57.2957795130823208768f
#define INV_SQRT2 0.70710678118654752440f

typedef float v2f __attribute__((ext_vector_type(2)));
typedef float v8f __attribute__((ext_vector_type(8)));

#if defined(__has_builtin)
#if __has_builtin(__builtin_amdgcn_global_load_async_to_lds_b32)
#define USE_ASYNC_LDS 1
#endif
#endif

#if defined(USE_ASYNC_LDS)
typedef __attribute__((address_space(1))) int* glb_i32_ptr;
typedef __attribute__((address_space(3))) int* lds_i32_ptr;

__device__ __forceinline__ void wait_async_zero() {
#if __has_builtin(__builtin_amdgcn_s_wait_asynccnt)
  __builtin_amdgcn_s_wait_asynccnt(0);
#else
  asm volatile("s_wait_asynccnt 0" ::: "memory");
#endif
}
#endif

// Sum 128 floats living in LDS using two chained V_WMMA_F32_16X16X4_F32.
// A-matrix (16x4 f32) lane layout per CDNA5 ISA: lane L holds
// A[L&15][k0] in VGPR0 and A[L&15][k0+1] in VGPR1, k0 = (L<16) ? 0 : 2.
// We map element s[4*m + k] -> A[m][k]; with B == ones, D[m][n] = sum_k A[m][k] + C.
// After two WMMAs (C-accumulated), summing the 8 D VGPRs in a lane gives
// sum over M=0..7 (lanes 0-15) / M=8..15 (lanes 16-31); one xor-16 shuffle
// combines the halves. All 32 lanes end with the full 128-element sum.
__device__ __forceinline__ float wmma_reduce128(const float* __restrict__ s,
                                                unsigned lane) {
  const unsigned base = ((lane & 15u) << 2) | ((lane >> 4) << 1);
  v2f a0, a1, b;
  a0[0] = s[base];
  a0[1] = s[base + 1];
  a1[0] = s[64 + base];
  a1[1] = s[64 + base + 1];
  b[0] = 1.0f;
  b[1] = 1.0f;
  v8f c = {0.f, 0.f, 0.f, 0.f, 0.f, 0.f, 0.f, 0.f};
  c = __builtin_amdgcn_wmma_f32_16x16x4_f32(false, a0, false, b, (short)0, c,
                                            false, false);
  c = __builtin_amdgcn_wmma_f32_16x16x4_f32(false, a1, false, b, (short)0, c,
                                            false, false);
  float h = c[0] + c[1] + c[2] + c[3] + c[4] + c[5] + c[6] + c[7];
  h += __shfl_xor(h, 16, 32);
  return h;
}

__global__ void __launch_bounds__(BLK) mt_forward_kernel(
    const float* __restrict__ rho, const float* __restrict__ thick,
    const float* __restrict__ freq, const float* __restrict__ obs_rhoa,
    const float* __restrict__ obs_phase, float* __restrict__ partial) {
  // Frequency-independent per-layer terms:
  //   sZ[j] = sqrt(MU * rho_j)          (impedance magnitude coefficient)
  //   sA[j] = sqrt(2*MU / rho_j) * t_j  (attenuation coefficient: a = sw * sA)
  __shared__ float sZ[NZc];
  __shared__ float sA[NZc];
  __shared__ float sRed[2][BLK];
  const unsigned tid = threadIdx.x;

#if defined(USE_ASYNC_LDS)
  __shared__ float sRaw[2 * NZc];
  for (unsigned z = tid; z < NZc; z += BLK) {
    __builtin_amdgcn_global_load_async_to_lds_b32(
        (glb_i32_ptr)(rho + z), (lds_i32_ptr)(&sRaw[z]), 0, 0);
  }
  for (unsigned z = tid; z < NZc - 1; z += BLK) {
    __builtin_amdgcn_global_load_async_to_lds_b32(
        (glb_i32_ptr)(thick + z), (lds_i32_ptr)(&sRaw[NZc + z]), 0, 0);
  }
  wait_async_zero();
  __syncthreads();
  for (unsigned z = tid; z < NZc; z += BLK) {
    const float r = sRaw[z];
    sZ[z] = sqrtf(MUc * r);
    sA[z] = (z < NZc - 1) ? sqrtf(2.0f * MUc / r) * sRaw[NZc + z] : 0.0f;
  }
#else
  for (unsigned z = tid; z < NZc; z += BLK) {
    const float r = rho[z];
    sZ[z] = sqrtf(MUc * r);
    sA[z] = (z < NZc - 1) ? sqrtf(2.0f * MUc / r) * thick[z] : 0.0f;
  }
#endif
  __syncthreads();

  const unsigned i = blockIdx.x * BLK + tid;  // one frequency per thread
  const float f = freq[i];
  const float omega = TWO_PI * f;
  const float sw = sqrtf(omega);

  // Z0 = sqrt(omega*MU*rho[NZ-1]) * (1+i)/sqrt(2)
  const float m0 = sw * sZ[NZc - 1] * INV_SQRT2;
  float Zr = m0, Zi = m0;

  // Propagate from layer NZ-2 down to 0 (matches reversed lax.scan order).
#pragma unroll 4
  for (int j = NZc - 2; j >= 0; --j) {
    const float Zjm = sw * sZ[j] * INV_SQRT2;  // Zj = Zjm*(1 + i)
    // r = (Z - Zj) / (Z + Zj)
    const float nr = Zr - Zjm, ni = Zi - Zjm;
    const float dr = Zr + Zjm, di = Zi + Zjm;
    const float inv1 = 1.0f / (dr * dr + di * di);
    const float rr = (nr * dr + ni * di) * inv1;
    const float ri = (ni * dr - nr * di) * inv1;
    // exp(-2*k*t) = exp(-a) * (cos a - i sin a), a = sw * sA[j]
    const float a = sw * sA[j];
    const float e = expf(-a);
    float sa, ca;
    sincosf(a, &sa, &ca);
    const float er = e * ca, ei = -e * sa;
    // re = r * exp(-2kt)
    const float rer = rr * er - ri * ei;
    const float rei = rr * ei + ri * er;
    // q = (1 - re) / (1 + re)
    const float qnr = 1.0f - rer, qni = -rei;
    const float qdr = 1.0f + rer, qdi = rei;
    const float inv2 = 1.0f / (qdr * qdr + qdi * qdi);
    const float qr = (qnr * qdr + qni * qdi) * inv2;
    const float qi = (qni * qdr - qnr * qdi) * inv2;
    // Z = Zj * q = Zjm*(1+i)*(qr + i qi)
    Zr = Zjm * (qr - qi);
    Zi = Zjm * (qr + qi);
  }

  const float om_mu = omega * MUc;
  const float app_res = (Zr * Zr + Zi * Zi) / om_mu;
  const float dlr = log10f(app_res) - log10f(obs_rhoa[i]);
  const float phase = atan2f(Zi, Zr) * RAD2DEG;
  const float dph = phase - obs_phase[i];

  sRed[0][tid] = dlr * dlr;
  sRed[1][tid] = dph * dph;
  __syncthreads();

  // All 4 waves run the identical WMMA reduction (EXEC all-ones everywhere);
  // thread 0 publishes the block partials.
  const unsigned lane = tid & 31u;
  const float s1 = wmma_reduce128(sRed[0], lane);
  const float s2 = wmma_reduce128(sRed[1], lane);
  if (tid == 0) {
    partial[2u * blockIdx.x] = s1;
    partial[2u * blockIdx.x + 1u] = s2;
  }
}

__global__ void __launch_bounds__(BLK) mt_finalize_kernel(
    const float* __restrict__ partial, float* __restrict__ out) {
  __shared__ float sRed[2][BLK];
  const unsigned tid = threadIdx.x;  // 128 threads, 128 block-partials
  sRed[0][tid] = partial[2u * tid];
  sRed[1][tid] = partial[2u * tid + 1u];
  __syncthreads();
  const unsigned lane = tid & 31u;
  const float s1 = wmma_reduce128(sRed[0], lane);
  const float s2 = wmma_reduce128(sRed[1], lane);
  if (tid == 0) {
    const float loss_rhoa = s1 * (1.0f / (float)NFc);
    const float loss_phase = s2 * (1.0f / (float)NFc);
    out[0] = loss_rhoa + 10.0f * loss_phase;  // total (LAMBDA_RHOA=1, LAMBDA_PHASE=10)
    out[1] = loss_rhoa;
    out[2] = loss_phase;
  }
}

extern "C" void kernel_launch(void* const* d_in, const int* in_sizes, int n_in,
                              void* d_out, int out_size, void* d_ws,
                              size_t ws_size, hipStream_t stream) {
  const float* rho = (const float*)d_in[0];        // (256,)
  const float* thick = (const float*)d_in[1];      // (255,)
  const float* freq = (const float*)d_in[2];       // (16384,)
  const float* obs_rhoa = (const float*)d_in[3];   // (16384,)
  const float* obs_phase = (const float*)d_in[4];  // (16384,)
  float* out = (float*)d_out;                      // 3 floats
  float* partial = (float*)d_ws;                   // 2 * 128 floats

  mt_forward_kernel<<<NFc / BLK, BLK, 0, stream>>>(rho, thick, freq, obs_rhoa,
                                                   obs_phase, partial);
  mt_finalize_kernel<<<1, BLK, 0, stream>>>(partial, out);
}